// SpatialTemporal_MS_GCN_33792802685293
// MI455X (gfx1250) — compile-verified
//
#include <hip/hip_runtime.h>
#include <hip/hip_bf16.h>

// Problem constants (from reference setup_inputs)
#define N_  32
#define C_  96
#define T_  300
#define V_  25
#define O_  192
#define TPB 4          // t-values per block
#define TV  (T_ * V_)  // 7500

// LDS strides chosen for bank-conflict-free WMMA fragment reads (64 banks)
constexpr int WSTR = 100;  // 100*4B: 36*row mod 64 permutes -> conflict-free A-frags
constexpr int YSTR = 26;
constexpr int ASTR = 26;
constexpr int GSTR = 48;   // 2*48 = 96 ≡ 32 (mod 64): half-waves hit disjoint banks

typedef __attribute__((ext_vector_type(2))) float v2f;
typedef __attribute__((ext_vector_type(8))) float v8f;

#if defined(__HIP_DEVICE_COMPILE__) && !__has_builtin(__builtin_amdgcn_wmma_f32_16x16x4_f32)
#error "missing __builtin_amdgcn_wmma_f32_16x16x4_f32 on this toolchain (device pass)"
#endif

__global__ __launch_bounds__(256) void msgcn_fused_kernel(
    const float* __restrict__ x,      // [N,C,T,V]
    const float* __restrict__ xx,     // [N,3,T,V]
    const float* __restrict__ W,      // [O,C]
    const float* __restrict__ b,      // [O]
    const float* __restrict__ gamma,  // [O]
    const float* __restrict__ beta,   // [O]
    float* __restrict__ out)          // [N,O,T,V]
{
    __shared__ float Wl[O_ * WSTR];    // folded weights (gamma*inv_std pre-applied)
    __shared__ float biasl[O_];        // folded bias
    __shared__ float ysl[C_ * YSTR];   // x tile  [c][u]
    __shared__ float Anl[V_ * ASTR];   // normalized adjacency [v][u]
    __shared__ float dinvl[V_];
    __shared__ float aggl[C_ * GSTR];  // agg tile [c][v] (B-matrix for WMMA)
    __shared__ float cxl[V_ * 4];      // coords [v][xyz-]

    const int tid = threadIdx.x;
    const int n   = blockIdx.x / (T_ / TPB);
    const int t0  = (blockIdx.x % (T_ / TPB)) * TPB;

    const float inv_std = 0.99999500003749975f;  // 1/sqrt(1 + 1e-5)

    // Stage folded weights: W'[o][c] = W[o,c] * gamma[o] * inv_std
    for (int idx = tid; idx < O_ * C_; idx += 256) {
        const int o = idx / C_, c = idx - o * C_;
        Wl[o * WSTR + c] = W[idx] * (gamma[o] * inv_std);
    }
    if (tid < O_) biasl[tid] = gamma[tid] * inv_std * b[tid] + beta[tid];
    // Zero agg padding columns 25..31 (read by N-tile 1 B-fragments, never stored)
    for (int idx = tid; idx < C_ * 7; idx += 256) {
        const int c = idx / 7, v = 25 + (idx - c * 7);
        aggl[c * GSTR + v] = 0.0f;
    }

    // Lane decomposition for WMMA fragments
    const int lane = tid & 31;
    const int wv   = tid >> 5;       // wave 0..7
    const int ntile = wv & 1;        // which N-tile (v columns 0..15 / 16..31)
    const int mg    = wv >> 1;       // M-group 0..3 (3 M-tiles each)
    const int lrow  = lane & 15;
    const int lhi   = lane >> 4;     // half-wave select
    const int nb    = ntile * 16;

    for (int tt = 0; tt < TPB; ++tt) {
        const int t = t0 + tt;
        __syncthreads();  // previous iteration's readers are done

        // ---- coords [v][xyz] ----
        if (tid < 3 * V_) {
            const int j = tid / V_, v = tid - j * V_;
            cxl[v * 4 + j] = xx[(n * 3 + j) * TV + t * V_ + v];
        }
        // ---- y tile: ysl[c][u] = x[n,c,t,u] (issued early, consumed after 4 barriers) ----
        const float* xp = x + (size_t)n * C_ * TV + (size_t)t * V_;
        for (int idx = tid; idx < C_ * V_; idx += 256) {
            const int c = idx / V_, u = idx - c * V_;
            ysl[c * YSTR + u] = xp[c * TV + u];
        }
        __syncthreads();

        // ---- A = exp(-||ci - cj||^2) ----
        for (int idx = tid; idx < V_ * V_; idx += 256) {
            const int v = idx / V_, u = idx - v * V_;
            const float dx = cxl[v * 4 + 0] - cxl[u * 4 + 0];
            const float dy = cxl[v * 4 + 1] - cxl[u * 4 + 1];
            const float dz = cxl[v * 4 + 2] - cxl[u * 4 + 2];
            Anl[v * ASTR + u] = __expf(-(dx * dx + dy * dy + dz * dz));
        }
        __syncthreads();

        // ---- degree^{-1/2} ----
        if (tid < V_) {
            float s = 0.0f;
            #pragma unroll
            for (int u = 0; u < V_; ++u) s += Anl[tid * ASTR + u];
            dinvl[tid] = rsqrtf(s);
        }
        __syncthreads();

        // ---- symmetric normalize ----
        for (int idx = tid; idx < V_ * V_; idx += 256) {
            const int v = idx / V_, u = idx - v * V_;
            Anl[v * ASTR + u] *= dinvl[v] * dinvl[u];
        }
        __syncthreads();

        // ---- graph aggregation (VALU): agg[c][v] = sum_u y[c][u] * An[v][u] ----
        for (int idx = tid; idx < C_ * V_; idx += 256) {
            const int c = idx / V_, v = idx - c * V_;
            float s = 0.0f;
            #pragma unroll
            for (int u = 0; u < V_; ++u)
                s = fmaf(ysl[c * YSTR + u], Anl[v * ASTR + u], s);
            aggl[c * GSTR + v] = s;
        }
        __syncthreads();

        // ---- MLP GEMM via V_WMMA_F32_16X16X4_F32: out[o][v] = W'[o][:] . agg[:][v] ----
        // Cache this wave's B fragments (agg, one N-tile, all K=96) in registers.
        v2f bf[24];
        #pragma unroll
        for (int ks = 0; ks < 24; ++ks) {
            const int kr = 4 * ks + 2 * lhi;
            bf[ks].x = aggl[kr * GSTR + nb + lrow];
            bf[ks].y = aggl[(kr + 1) * GSTR + nb + lrow];
        }

        float* outp = out + (size_t)n * O_ * TV + (size_t)t * V_;
        const int vcol = nb + lrow;

        #pragma unroll
        for (int mi = 0; mi < 3; ++mi) {
            const int mbase = (mg * 3 + mi) * 16;
            v8f acc = {0.f, 0.f, 0.f, 0.f, 0.f, 0.f, 0.f, 0.f};
            #pragma unroll
            for (int ks = 0; ks < 24; ++ks) {
                const int kc = 4 * ks + 2 * lhi;
                v2f af;
                af.x = Wl[(mbase + lrow) * WSTR + kc];
                af.y = Wl[(mbase + lrow) * WSTR + kc + 1];
                acc = __builtin_amdgcn_wmma_f32_16x16x4_f32(
                    false, af, false, bf[ks], (short)0, acc, false, false);
            }
            // fused bias + BN(affine) + ReLU store; mask padded v columns
            if (vcol < V_) {
                #pragma unroll
                for (int r = 0; r < 8; ++r) {
                    const int o = mbase + r + 8 * lhi;
                    const float val = fmaxf(acc[r] + biasl[o], 0.0f);
                    outp[o * TV + vcol] = val;
                }
            }
        }
    }
}

extern "C" void kernel_launch(void* const* d_in, const int* in_sizes, int n_in,
                              void* d_out, int out_size, void* d_ws, size_t ws_size,
                              hipStream_t stream) {
    const float* x     = (const float*)d_in[0];
    const float* xx    = (const float*)d_in[1];
    const float* W     = (const float*)d_in[2];
    const float* b     = (const float*)d_in[3];
    const float* gamma = (const float*)d_in[4];
    const float* beta  = (const float*)d_in[5];
    float* out = (float*)d_out;

    dim3 grid(N_ * (T_ / TPB));  // 32 * 75 = 2400 blocks, 8 waves each
    msgcn_fused_kernel<<<grid, 256, 0, stream>>>(x, xx, W, b, gamma, beta, out);
}